// GINEConvHetero_30227979829589
// MI455X (gfx1250) — compile-verified
//
#include <hip/hip_runtime.h>

typedef __attribute__((ext_vector_type(2))) float v2f;
typedef __attribute__((ext_vector_type(8))) float v8f;

#define NDIM 64

__device__ __forceinline__ v8f zero8() { v8f z = {}; return z; }

// Accumulate one 16x16 f32 tile: C += A[m0:m0+16, 0:64] * W[wrow0:wrow0+64, n0:n0+16]
// A is an LDS tile (32 rows x 64 cols, row-major). W is a global row-major matrix (64 cols).
__device__ __forceinline__ v8f wmma_acc(v8f c,
                                        const float* __restrict__ A,
                                        const float* __restrict__ W, int wrow0,
                                        int m0, int n0, int half, int lm)
{
  const float* arow = A + (m0 + lm) * NDIM;
  const float* wcol = W + (size_t)wrow0 * NDIM + n0 + lm;
#pragma unroll
  for (int k0 = 0; k0 < NDIM; k0 += 4) {
    const int k = k0 + 2 * half;
    v2f a, b;
    a.x = arow[k];
    a.y = arow[k + 1];
    b.x = wcol[(size_t)k * NDIM];
    b.y = wcol[(size_t)(k + 1) * NDIM];
    // D = A*B + C  (fp32 matrix pipe, 16x16x4)
    c = __builtin_amdgcn_wmma_f32_16x16x4_f32(false, a, false, b, (short)0, c,
                                              false, false);
  }
  return c;
}

__device__ __forceinline__ void store_tile(float* __restrict__ S, int m0, int n0,
                                           int half, int lm, v8f c)
{
#pragma unroll
  for (int v = 0; v < 8; ++v)
    S[(m0 + 8 * half + v) * NDIM + n0 + lm] = c[v];
}

// ---------------------------------------------------------------------------
// Kernel 0: zero the aggregation workspace (2 * N * 64 floats)
// ---------------------------------------------------------------------------
__global__ __launch_bounds__(256) void gine_zero_ws(float4* __restrict__ p, long long n4)
{
  long long i = (long long)blockIdx.x * blockDim.x + threadIdx.x;
  const long long stride = (long long)gridDim.x * blockDim.x;
  float4 z; z.x = z.y = z.z = z.w = 0.0f;
  for (; i < n4; i += stride) p[i] = z;
}

// ---------------------------------------------------------------------------
// Kernel 1: edge scatter.  16 threads per edge, one float4 chunk each.
//   agg_in[dst]  += relu(x[src] + ea)
//   agg_out[src] += relu(x[dst] + ea)
// ---------------------------------------------------------------------------
__global__ __launch_bounds__(256) void gine_edge_scatter(
    const int* __restrict__ src, const int* __restrict__ dst,
    const float* __restrict__ x, const float* __restrict__ ea,
    float* __restrict__ agg_in, float* __restrict__ agg_out, int E)
{
  const long long gid = (long long)blockIdx.x * blockDim.x + threadIdx.x;
  const int e = (int)(gid >> 4);
  if (e >= E) return;
  const int c = (int)(gid & 15);

  const int s = src[e];
  const int d = dst[e];

  const float4* x4 = (const float4*)x;
  const float4 a4 = ((const float4*)ea)[(size_t)e * 16 + c];
  const float4 xs = x4[(size_t)s * 16 + c];
  const float4 xd = x4[(size_t)d * 16 + c];

  float f0 = fmaxf(xs.x + a4.x, 0.0f);
  float f1 = fmaxf(xs.y + a4.y, 0.0f);
  float f2 = fmaxf(xs.z + a4.z, 0.0f);
  float f3 = fmaxf(xs.w + a4.w, 0.0f);
  float g0 = fmaxf(xd.x + a4.x, 0.0f);
  float g1 = fmaxf(xd.y + a4.y, 0.0f);
  float g2 = fmaxf(xd.z + a4.z, 0.0f);
  float g3 = fmaxf(xd.w + a4.w, 0.0f);

  float* pi = agg_in  + (size_t)d * NDIM + c * 4;
  float* po = agg_out + (size_t)s * NDIM + c * 4;
  __hip_atomic_fetch_add(pi + 0, f0, __ATOMIC_RELAXED, __HIP_MEMORY_SCOPE_AGENT);
  __hip_atomic_fetch_add(pi + 1, f1, __ATOMIC_RELAXED, __HIP_MEMORY_SCOPE_AGENT);
  __hip_atomic_fetch_add(pi + 2, f2, __ATOMIC_RELAXED, __HIP_MEMORY_SCOPE_AGENT);
  __hip_atomic_fetch_add(pi + 3, f3, __ATOMIC_RELAXED, __HIP_MEMORY_SCOPE_AGENT);
  __hip_atomic_fetch_add(po + 0, g0, __ATOMIC_RELAXED, __HIP_MEMORY_SCOPE_AGENT);
  __hip_atomic_fetch_add(po + 1, g1, __ATOMIC_RELAXED, __HIP_MEMORY_SCOPE_AGENT);
  __hip_atomic_fetch_add(po + 2, g2, __ATOMIC_RELAXED, __HIP_MEMORY_SCOPE_AGENT);
  __hip_atomic_fetch_add(po + 3, g3, __ATOMIC_RELAXED, __HIP_MEMORY_SCOPE_AGENT);
}

// ---------------------------------------------------------------------------
// Kernel 2: fused MLP(agg_in), MLP(agg_out), and final projection.
// One block = 256 threads = 8 waves = (2 M-tiles x 4 N-tiles) = 32 nodes.
// All GEMMs via v_wmma_f32_16x16x4_f32.
// ---------------------------------------------------------------------------
__global__ __launch_bounds__(256) void gine_mlp_proj(
    const float* __restrict__ x,
    const float* __restrict__ gin,
    const float* __restrict__ gout,
    const float* __restrict__ W1, const float* __restrict__ b1,
    const float* __restrict__ W2, const float* __restrict__ b2,
    const float* __restrict__ Wl, const float* __restrict__ bl,
    float* __restrict__ out, int N)
{
  __shared__ __align__(16) float s_x[32 * NDIM];
  __shared__ __align__(16) float s_gin[32 * NDIM];
  __shared__ __align__(16) float s_gout[32 * NDIM];
  __shared__ __align__(16) float s_h[32 * NDIM];
  __shared__ __align__(16) float s_ain[32 * NDIM];
  __shared__ __align__(16) float s_aout[32 * NDIM];

  const int tid = threadIdx.x;
  const int row0 = blockIdx.x * 32;
  int rows = N - row0;
  if (rows > 32) rows = 32;

  // Cooperative coalesced tile loads (512 float4 per tile / 256 threads).
  {
    const float4* gx = (const float4*)(x    + (size_t)row0 * NDIM);
    const float4* gi = (const float4*)(gin  + (size_t)row0 * NDIM);
    const float4* go = (const float4*)(gout + (size_t)row0 * NDIM);
    float4 z4; z4.x = z4.y = z4.z = z4.w = 0.0f;
    for (int i = tid; i < 32 * (NDIM / 4); i += 256) {
      const int r = i >> 4;
      const bool ok = (r < rows);
      ((float4*)s_x)[i]    = ok ? gx[i] : z4;
      ((float4*)s_gin)[i]  = ok ? gi[i] : z4;
      ((float4*)s_gout)[i] = ok ? go[i] : z4;
    }
  }
  __syncthreads();

  const int lane = tid & 31;
  const int wave = tid >> 5;
  const int half = lane >> 4;
  const int lm   = lane & 15;
  const int m0   = (wave >> 2) * 16;  // 0 or 16
  const int n0   = (wave & 3) * 16;   // 0,16,32,48

  v8f c;
  float bcol;

  // ---- branch IN: a_in = relu(gin@W1 + b1) @ W2 + b2 ----
  c = zero8();
  c = wmma_acc(c, s_gin, W1, 0, m0, n0, half, lm);
  bcol = b1[n0 + lm];
#pragma unroll
  for (int v = 0; v < 8; ++v) c[v] = fmaxf(c[v] + bcol, 0.0f);
  store_tile(s_h, m0, n0, half, lm, c);
  __syncthreads();

  c = zero8();
  c = wmma_acc(c, s_h, W2, 0, m0, n0, half, lm);
  bcol = b2[n0 + lm];
#pragma unroll
  for (int v = 0; v < 8; ++v) c[v] += bcol;
  store_tile(s_ain, m0, n0, half, lm, c);
  __syncthreads();

  // ---- branch OUT: a_out = relu(gout@W1 + b1) @ W2 + b2 ----
  c = zero8();
  c = wmma_acc(c, s_gout, W1, 0, m0, n0, half, lm);
  bcol = b1[n0 + lm];
#pragma unroll
  for (int v = 0; v < 8; ++v) c[v] = fmaxf(c[v] + bcol, 0.0f);
  store_tile(s_h, m0, n0, half, lm, c);
  __syncthreads();

  c = zero8();
  c = wmma_acc(c, s_h, W2, 0, m0, n0, half, lm);
  bcol = b2[n0 + lm];
#pragma unroll
  for (int v = 0; v < 8; ++v) c[v] += bcol;
  store_tile(s_aout, m0, n0, half, lm, c);
  __syncthreads();

  // ---- final: out = x@Wl[0:64] + a_in@Wl[64:128] + a_out@Wl[128:192] + bl ----
  c = zero8();
  c = wmma_acc(c, s_x,    Wl, 0,   m0, n0, half, lm);
  c = wmma_acc(c, s_ain,  Wl, 64,  m0, n0, half, lm);
  c = wmma_acc(c, s_aout, Wl, 128, m0, n0, half, lm);
  bcol = bl[n0 + lm];
#pragma unroll
  for (int v = 0; v < 8; ++v) {
    const int r = m0 + 8 * half + v;
    if (r < rows)
      out[(size_t)(row0 + r) * NDIM + n0 + lm] = c[v] + bcol;
  }
}

// ---------------------------------------------------------------------------
extern "C" void kernel_launch(void* const* d_in, const int* in_sizes, int n_in,
                              void* d_out, int out_size, void* d_ws, size_t ws_size,
                              hipStream_t stream)
{
  (void)n_in; (void)out_size; (void)ws_size;
  const float* x  = (const float*)d_in[0];
  const int*   ei = (const int*)d_in[1];
  const float* ea = (const float*)d_in[2];
  const float* W1 = (const float*)d_in[3];
  const float* b1 = (const float*)d_in[4];
  const float* W2 = (const float*)d_in[5];
  const float* b2 = (const float*)d_in[6];
  const float* Wl = (const float*)d_in[7];
  const float* bl = (const float*)d_in[8];
  float* out = (float*)d_out;

  const int N = in_sizes[0] / NDIM;
  const int E = in_sizes[1] / 2;
  const int* src = ei;       // edge_index[0]
  const int* dst = ei + E;   // edge_index[1]

  float* agg_in  = (float*)d_ws;
  float* agg_out = agg_in + (size_t)N * NDIM;

  // 0) zero aggregation buffers
  const long long n4 = (2LL * N * NDIM) / 4;
  gine_zero_ws<<<4096, 256, 0, stream>>>((float4*)d_ws, n4);

  // 1) edge scatter (16 threads per edge)
  const long long sthreads = (long long)E * 16;
  const int sblocks = (int)((sthreads + 255) / 256);
  gine_edge_scatter<<<sblocks, 256, 0, stream>>>(src, dst, x, ea, agg_in, agg_out, E);

  // 2) fused MLPs + projection (WMMA), 32 nodes per block
  const int mblocks = (N + 31) / 32;
  gine_mlp_proj<<<mblocks, 256, 0, stream>>>(x, agg_in, agg_out,
                                             W1, b1, W2, b2, Wl, bl, out, N);
}